// HGCN_90305982365735
// MI455X (gfx1250) — compile-verified
//
#include <hip/hip_runtime.h>
#include <hip/hip_bf16.h>

// ---------------------------------------------------------------------------
// RGCN encoder + bilinear-diag decoder for MI455X (gfx1250, wave32, WMMA).
//
//   k_zero       : zero cnt[R*N]
//   k_relW       : W[r] = comp[r] @ basis          (32x32x8, tiny)
//   k_encoder    : h = x@emb_W + b  (WMMA f32 16x16x4, LDS-staged x tiles,
//                  k-pair-interleaved B layout, 32 rows/wave, 4 C tiles)
//                  + epilogue: acc = h@root_W + rgcn_bias (2nd WMMA pass)
//   k_count      : cnt[et*N+dst]++ per edge
//   k_message    : per edge: msg = h[src]@W_lds[et]; acc[dst] += msg/cnt
//   k_relu       : z = relu(acc)
//   k_decode     : pos/neg sigmoid(z[s]*rel[et]*z[d]) -> d_out[0:E], [E:2E]
// ---------------------------------------------------------------------------

#define NN      100000
#define IN_DIM  512
#define EMB     32
#define OUT     8
#define NUM_ET  32
#define NUM_BASE 16

typedef float v2f __attribute__((ext_vector_type(2)));
typedef float v8f __attribute__((ext_vector_type(8)));

__device__ __forceinline__ v8f wmma_f32_4(v2f a, v2f b, v8f c) {
    // V_WMMA_F32_16X16X4_F32 : D = A(16x4) * B(4x16) + C(16x16)
    return __builtin_amdgcn_wmma_f32_16x16x4_f32(
        /*neg_a=*/false, a, /*neg_b=*/false, b,
        /*c_mod=*/(short)0, c, /*reuse_a=*/false, /*reuse_b=*/false);
}

// ---------------------------------------------------------------------------
__global__ void k_zero_u32(unsigned* __restrict__ p, int n) {
    int i = blockIdx.x * 256 + threadIdx.x;
    if (i < n) p[i] = 0u;
}

// W[r][i][o] = sum_b comp[r][b] * basis[b][i][o]   -> [32, 32, 8]
__global__ void k_relW(const float* __restrict__ comp,
                       const float* __restrict__ basis,
                       float* __restrict__ W) {
    int idx = blockIdx.x * 256 + threadIdx.x;          // 32*256 total
    if (idx >= NUM_ET * EMB * OUT) return;
    int r  = idx >> 8;                                  // /256
    int io = idx & 255;
    float s = 0.f;
#pragma unroll
    for (int b = 0; b < NUM_BASE; ++b)
        s += comp[r * NUM_BASE + b] * basis[b * (EMB * OUT) + io];
    W[idx] = s;
}

// ---------------------------------------------------------------------------
// Encoder GEMM: h[N,32] = x[N,512] @ embW[512,32] + embB
// Block = 128 threads (4 waves), 128 rows/block (32 rows/wave, 4 C tiles).
// B staged k-pair interleaved: BtileP[kp][n][2] so a lane's {B[k][n],B[k+1][n]}
// is one contiguous b64, and b0/b1 (n, n+16) come from one ds_load_2addr_b64.
// Epilogue: acc[N,8] = h @ rootW[32,8 padded] + rgcn_bias  (2nd WMMA pass)
// ---------------------------------------------------------------------------
#define AT_STRIDE 68   // LDS row stride (floats): 8B-aligned, bank-conflict free
#define BROWS 128
__global__ __launch_bounds__(128) void k_encoder(
    const float* __restrict__ x,    const float* __restrict__ embW,
    const float* __restrict__ embB, const float* __restrict__ rootW,
    const float* __restrict__ rgcnB,
    float* __restrict__ h, float* __restrict__ acc)
{
    __shared__ float Atile[BROWS * AT_STRIDE];   // 34816 B
    __shared__ float BtileP[64 * 32];            //  8192 B, [kp][n][2]

    const int tid  = threadIdx.x;
    const int wave = tid >> 5;
    const int lane = tid & 31;
    const int l16  = lane & 15;
    const int hi   = lane >> 4;              // 0: K+0/1, 1: K+2/3
    const int rowBase = blockIdx.x * BROWS;

    v8f c00 = {}, c01 = {}, c10 = {}, c11 = {};

    for (int kc = 0; kc < IN_DIM; kc += 64) {
        // ---- cooperative, coalesced stage of x[rowBase..+127][kc..+63]
        {
            const int col4 = (tid & 15) * 4;
            const int r0   = tid >> 4;                 // 0..7
#pragma unroll
            for (int i = 0; i < 16; ++i) {
                int r  = r0 + i * 8;
                int rg = rowBase + r; if (rg >= NN) rg = NN - 1;  // clamp tail
                const float* gp = x + (size_t)rg * IN_DIM + kc + col4;
                float4 v = *(const float4*)gp;
                if (i == 0 && kc + 64 < IN_DIM)
                    __builtin_prefetch(gp + 64, 0, 0);  // next K-chunk
                *(float4*)(Atile + r * AT_STRIDE + col4) = v;
            }
            // embW rows kc..kc+63, interleaved by k-pairs
            const int bc4 = (tid & 7) * 4;             // 0..28
            const int br0 = tid >> 3;                  // 0..15
#pragma unroll
            for (int i = 0; i < 4; ++i) {
                int br = br0 + i * 16;                 // k row in chunk
                float4 v = *(const float4*)(embW + (size_t)(kc + br) * 32 + bc4);
                float* dstp = BtileP + (br >> 1) * 64 + (br & 1);
                dstp[(bc4 + 0) * 2] = v.x;
                dstp[(bc4 + 1) * 2] = v.y;
                dstp[(bc4 + 2) * 2] = v.z;
                dstp[(bc4 + 3) * 2] = v.w;
            }
        }
        __syncthreads();

        // ---- 16 k-steps of 4 : four 16x16 output tiles per wave
        const int ar0 = (wave * 32 + l16) * AT_STRIDE;
        const int ar1 = (wave * 32 + 16 + l16) * AT_STRIDE;
#pragma unroll
        for (int kk = 0; kk < 16; ++kk) {
            const int kof = kk * 4 + hi * 2;
            const int bo  = (kk * 2 + hi) * 64 + l16 * 2;
            v2f a0 = *(const v2f*)(Atile + ar0 + kof);
            v2f a1 = *(const v2f*)(Atile + ar1 + kof);
            v2f b0 = *(const v2f*)(BtileP + bo);        // cols 0-15
            v2f b1 = *(const v2f*)(BtileP + bo + 32);   // cols 16-31
            c00 = wmma_f32_4(a0, b0, c00);
            c01 = wmma_f32_4(a0, b1, c01);
            c10 = wmma_f32_4(a1, b0, c10);
            c11 = wmma_f32_4(a1, b1, c11);
        }
        __syncthreads();
    }

    // ---- bias, store h, stage tile for root GEMM (wave-private 32x32)
    const float bb0 = embB[l16];
    const float bb1 = embB[16 + l16];
    float* hT = Atile + wave * 32 * AT_STRIDE;
#pragma unroll
    for (int j = 0; j < 8; ++j) {
        const int m  = j + hi * 8;                       // C layout: M=j+8*hi
        {   // rows 0-15 of wave tile
            const int rg = rowBase + wave * 32 + m;
            const float v0 = c00[j] + bb0;
            const float v1 = c01[j] + bb1;
            hT[m * AT_STRIDE + l16]      = v0;
            hT[m * AT_STRIDE + 16 + l16] = v1;
            if (rg < NN) {
                h[(size_t)rg * EMB + l16]      = v0;
                h[(size_t)rg * EMB + 16 + l16] = v1;
            }
        }
        {   // rows 16-31 of wave tile
            const int m2 = 16 + m;
            const int rg = rowBase + wave * 32 + m2;
            const float v0 = c10[j] + bb0;
            const float v1 = c11[j] + bb1;
            hT[m2 * AT_STRIDE + l16]      = v0;
            hT[m2 * AT_STRIDE + 16 + l16] = v1;
            if (rg < NN) {
                h[(size_t)rg * EMB + l16]      = v0;
                h[(size_t)rg * EMB + 16 + l16] = v1;
            }
        }
    }
    __syncthreads();

    // ---- root term: hT[32x32] @ rootW[32x8 padded to 16]  (WMMA pass 2)
    v8f c2a = {}, c2b = {};
#pragma unroll
    for (int kk = 0; kk < 8; ++kk) {
        const int kof = kk * 4 + hi * 2;
        v2f a0 = *(const v2f*)(hT + l16 * AT_STRIDE + kof);
        v2f a1 = *(const v2f*)(hT + (16 + l16) * AT_STRIDE + kof);
        v2f b;
        b.x = (l16 < OUT) ? rootW[kof * OUT + l16]       : 0.f;
        b.y = (l16 < OUT) ? rootW[(kof + 1) * OUT + l16] : 0.f;
        c2a = wmma_f32_4(a0, b, c2a);
        c2b = wmma_f32_4(a1, b, c2b);
    }
    if (l16 < OUT) {
        const float bias = rgcnB[l16];
#pragma unroll
        for (int j = 0; j < 8; ++j) {
            const int m   = j + hi * 8;
            const int rgA = rowBase + wave * 32 + m;
            const int rgB = rgA + 16;
            if (rgA < NN) acc[(size_t)rgA * OUT + l16] = c2a[j] + bias;
            if (rgB < NN) acc[(size_t)rgB * OUT + l16] = c2b[j] + bias;
        }
    }
}

// ---------------------------------------------------------------------------
__global__ __launch_bounds__(256) void k_count(
    const int* __restrict__ ei, const int* __restrict__ et,
    unsigned* __restrict__ cnt, int E)
{
    int e = blockIdx.x * 256 + threadIdx.x;
    if (e >= E) return;
    atomicAdd(&cnt[(size_t)et[e] * NN + ei[E + e]], 1u);
}

// per edge: msg = h[src] @ W[et];  acc[dst] += msg / cnt[et,dst]
__global__ __launch_bounds__(256) void k_message(
    const int* __restrict__ ei, const int* __restrict__ et,
    const float* __restrict__ h, const float* __restrict__ W,
    const unsigned* __restrict__ cnt, float* __restrict__ acc, int E)
{
    __shared__ float Wl[NUM_ET * EMB * OUT];            // 32 KB
    for (int i = threadIdx.x; i < NUM_ET * EMB * OUT; i += 256) Wl[i] = W[i];
    __syncthreads();

    int e = blockIdx.x * 256 + threadIdx.x;
    if (e >= E) return;
    const int s = ei[e];
    const int d = ei[E + e];
    const int r = et[e];
    const float* hs = h + (size_t)s * EMB;              // L2-resident (12.8 MB)
    const float* wr = Wl + r * (EMB * OUT);             // et sorted -> broadcast

    float msg[OUT] = {};
#pragma unroll
    for (int i = 0; i < EMB; ++i) {
        const float hv = hs[i];
#pragma unroll
        for (int o = 0; o < OUT; ++o) msg[o] += hv * wr[i * OUT + o];
    }
    const float inv = 1.f / fmaxf((float)cnt[(size_t)r * NN + d], 1.f);
    float* ap = acc + (size_t)d * OUT;
#pragma unroll
    for (int o = 0; o < OUT; ++o) atomicAdd(ap + o, msg[o] * inv);
}

__global__ void k_relu(const float* __restrict__ acc, float* __restrict__ z, int n) {
    int i = blockIdx.x * 256 + threadIdx.x;
    if (i < n) z[i] = fmaxf(acc[i], 0.f);
}

__global__ __launch_bounds__(256) void k_decode(
    const int* __restrict__ ei, const int* __restrict__ nei,
    const int* __restrict__ et, const float* __restrict__ z,
    const float* __restrict__ relE, float* __restrict__ out, int E)
{
    int e = blockIdx.x * 256 + threadIdx.x;
    if (e >= E) return;
    const int r = et[e];
    float rel[OUT];
#pragma unroll
    for (int o = 0; o < OUT; ++o) rel[o] = relE[r * OUT + o];

    const int s = ei[e], d = ei[E + e];
    float a1 = 0.f;
#pragma unroll
    for (int o = 0; o < OUT; ++o)
        a1 += z[(size_t)s * OUT + o] * rel[o] * z[(size_t)d * OUT + o];
    out[e] = 1.f / (1.f + __expf(-a1));

    const int ns = nei[e], nd = nei[E + e];
    float a2 = 0.f;
#pragma unroll
    for (int o = 0; o < OUT; ++o)
        a2 += z[(size_t)ns * OUT + o] * rel[o] * z[(size_t)nd * OUT + o];
    out[E + e] = 1.f / (1.f + __expf(-a2));
}

// ---------------------------------------------------------------------------
extern "C" void kernel_launch(void* const* d_in, const int* in_sizes, int n_in,
                              void* d_out, int out_size, void* d_ws, size_t ws_size,
                              hipStream_t stream) {
    const float* x     = (const float*)d_in[0];
    const int*   ei    = (const int*)  d_in[1];
    const int*   et    = (const int*)  d_in[2];
    /* d_in[3] = edge_type_num (unused: we count on device) */
    const int*   nei   = (const int*)  d_in[4];
    const float* embW  = (const float*)d_in[5];
    const float* embB  = (const float*)d_in[6];
    const float* basis = (const float*)d_in[7];
    const float* comp  = (const float*)d_in[8];
    const float* rootW = (const float*)d_in[9];
    const float* rgcnB = (const float*)d_in[10];
    const float* relE  = (const float*)d_in[11];
    float* out = (float*)d_out;
    const int E = in_sizes[2];

    // workspace layout (256B-aligned offsets), total ~32 MB
    char* ws = (char*)d_ws;
    float*    h   = (float*)(ws);                          // N*32  = 12.8 MB
    float*    acc = (float*)(ws + 12800000);               // N*8   =  3.2 MB
    float*    z   = (float*)(ws + 16000000);               // N*8   =  3.2 MB
    unsigned* cnt = (unsigned*)(ws + 19200000);            // R*N   = 12.8 MB
    float*    W   = (float*)(ws + 32000000);               // 32*256 = 32 KB

    const int RN = NUM_ET * NN;
    k_zero_u32<<<(RN + 255) / 256, 256, 0, stream>>>(cnt, RN);
    k_relW<<<(NUM_ET * EMB * OUT + 255) / 256, 256, 0, stream>>>(comp, basis, W);
    k_encoder<<<(NN + BROWS - 1) / BROWS, 128, 0, stream>>>(x, embW, embB, rootW, rgcnB, h, acc);
    k_count<<<(E + 255) / 256, 256, 0, stream>>>(ei, et, cnt, E);
    k_message<<<(E + 255) / 256, 256, 0, stream>>>(ei, et, h, W, cnt, acc, E);
    k_relu<<<(NN * OUT + 255) / 256, 256, 0, stream>>>(acc, z, NN * OUT);
    k_decode<<<(E + 255) / 256, 256, 0, stream>>>(ei, nei, et, z, relE, out, E);
}